// PoseLoss_29480655520234
// MI455X (gfx1250) — compile-verified
//
#include <hip/hip_runtime.h>
#include <math.h>

// ---------------------------------------------------------------------------
// PoseLoss for MI455X (gfx1250, wave32).
// 100 MB streamed -> 1 float. Jointly HBM / transcendental-VALU bound.
// - Stage 1: double-buffered ASYNC global->LDS staging (gfx1250
//   global_load_async_to_lds_b128 + s_wait_asynccnt): deep prefetch with no
//   VGPR destinations in flight, then per-lane 3x3 algebra with HW
//   transcendentals (v_rsq/v_sin/v_cos/v_sqrt), wave32 __shfl_xor reduction.
// - Stage 2: fold per-block partials deterministically, scale by 1/N.
// WMMA is intentionally NOT used: the matrix ops are 3x3 per lane (27 FMAs);
// staging them into 16x16 WMMA tiles moves more bytes than it saves math on a
// kernel that is not matrix-throughput bound.
// ---------------------------------------------------------------------------

#define PL_THREADS 256
#define PL_BLOCKS  1024

// ---- async tile issue: 6x b128 (48B pred + 48B gt per lane) ----------------
__device__ __forceinline__ void issue_tile_async(const float* __restrict__ pred,
                                                 const float* __restrict__ gt,
                                                 int j, unsigned ldsbase) {
  const float* pp = pred + (size_t)j * 12;  // pair j = items 2j,2j+1 (48B)
  const float* gg = gt   + (size_t)j * 12;
  // INST_OFFSET is applied to both the global address and the LDS address,
  // so one address VGPR pair serves all three 16B chunks.
  asm volatile("global_load_async_to_lds_b128 %0, %1, off"           :: "v"(ldsbase),       "v"(pp) : "memory");
  asm volatile("global_load_async_to_lds_b128 %0, %1, off offset:16" :: "v"(ldsbase),       "v"(pp) : "memory");
  asm volatile("global_load_async_to_lds_b128 %0, %1, off offset:32" :: "v"(ldsbase),       "v"(pp) : "memory");
  asm volatile("global_load_async_to_lds_b128 %0, %1, off"           :: "v"(ldsbase + 48u), "v"(gg) : "memory");
  asm volatile("global_load_async_to_lds_b128 %0, %1, off offset:16" :: "v"(ldsbase + 48u), "v"(gg) : "memory");
  asm volatile("global_load_async_to_lds_b128 %0, %1, off offset:32" :: "v"(ldsbase + 48u), "v"(gg) : "memory");
}

// ---- math helpers ----------------------------------------------------------

// Rodrigues: axis-angle -> 3x3 rotation (row-major R[3*r+c]).
__device__ __forceinline__ void aa_to_R(float ax, float ay, float az,
                                        float R[9]) {
  const float a2   = __fmaf_rn(ax, ax, __fmaf_rn(ay, ay, __fmaf_rn(az, az, 1e-20f)));
  const float rinv = rsqrtf(a2);       // v_rsq_f32 : 1/angle
  const float ang  = a2 * rinv;        // sqrt(a2) with one extra mul
  const float x = ax * rinv, y = ay * rinv, z = az * rinv;
  const float s = __sinf(ang);         // v_sin_f32
  const float c = __cosf(ang);         // v_cos_f32
  const float t = 1.0f - c;
  const float txy = t * x * y, txz = t * x * z, tyz = t * y * z;
  R[0] = 1.0f - t * (y * y + z * z);
  R[1] = txy - s * z;
  R[2] = txz + s * y;
  R[3] = txy + s * z;
  R[4] = 1.0f - t * (x * x + z * z);
  R[5] = tyz - s * x;
  R[6] = txz - s * y;
  R[7] = tyz + s * x;
  R[8] = 1.0f - t * (x * x + y * y);
}

// One pose pair: returns ||dr_aa|| + ||dt|| (per-item loss contribution).
__device__ __forceinline__ float pose_item(const float p[6], const float g[6]) {
  float P[9], G[9];
  aa_to_R(p[0], p[1], p[2], P);
  aa_to_R(g[0], g[1], g[2], G);

  // dr = G^T * P
  float dr[9];
#pragma unroll
  for (int i = 0; i < 3; ++i)
#pragma unroll
    for (int k = 0; k < 3; ++k)
      dr[3 * i + k] = __fmaf_rn(G[i], P[k],
                      __fmaf_rn(G[3 + i], P[3 + k], G[6 + i] * P[6 + k]));

  // dt = G^T * (tp - tg)
  const float d0 = p[3] - g[3], d1 = p[4] - g[4], d2 = p[5] - g[5];
  const float u0 = __fmaf_rn(G[0], d0, __fmaf_rn(G[3], d1, G[6] * d2));
  const float u1 = __fmaf_rn(G[1], d0, __fmaf_rn(G[4], d1, G[7] * d2));
  const float u2 = __fmaf_rn(G[2], d0, __fmaf_rn(G[5], d1, G[8] * d2));
  const float norm_t = __builtin_sqrtf(u0 * u0 + u1 * u1 + u2 * u2);

  // matrix -> quaternion (pytorch3d-stable variant)
  const float m00 = dr[0], m01 = dr[1], m02 = dr[2];
  const float m10 = dr[3], m11 = dr[4], m12 = dr[5];
  const float m20 = dr[6], m21 = dr[7], m22 = dr[8];

  const float qa0 = __builtin_sqrtf(fmaxf(0.0f, 1.0f + m00 + m11 + m22));
  const float qa1 = __builtin_sqrtf(fmaxf(0.0f, 1.0f + m00 - m11 - m22));
  const float qa2 = __builtin_sqrtf(fmaxf(0.0f, 1.0f - m00 + m11 - m22));
  const float qa3 = __builtin_sqrtf(fmaxf(0.0f, 1.0f - m00 - m11 + m22));

  // argmax (first max wins, matching jnp.argmax)
  int   idx  = 0;
  float best = qa0;
  if (qa1 > best) { best = qa1; idx = 1; }
  if (qa2 > best) { best = qa2; idx = 2; }
  if (qa3 > best) { best = qa3; idx = 3; }

  const float A = m21 - m12, B = m02 - m20, C = m10 - m01;
  const float D = m10 + m01, E = m02 + m20, F = m12 + m21;

  const float w  = (idx == 0) ? qa0 * qa0 : (idx == 1) ? A : (idx == 2) ? B : C;
  const float vx = (idx == 0) ? A : (idx == 1) ? qa1 * qa1 : (idx == 2) ? D : E;
  const float vy = (idx == 0) ? B : (idx == 1) ? D : (idx == 2) ? qa2 * qa2 : F;
  const float vz = (idx == 0) ? C : (idx == 1) ? E : (idx == 2) ? F : qa3 * qa3;

  const float sc = __fdividef(1.0f, 2.0f * fmaxf(best, 0.1f));
  const float qw = w * sc, qx = vx * sc, qy = vy * sc, qz = vz * sc;

  // quaternion -> axis-angle norm
  const float norms = __builtin_sqrtf(qx * qx + qy * qy + qz * qz + 1e-20f);
  const float half  = atan2f(norms, qw);
  const float angle = 2.0f * half;
  const bool  small = fabsf(angle) < 1e-6f;
  const float shoa  = small ? __fmaf_rn(angle * angle, -(1.0f / 48.0f), 0.5f)
                            : __fdividef(__sinf(half), angle);
  const float norm_r = __fdividef(norms, shoa);

  return norm_r + norm_t;
}

__device__ __forceinline__ float pose_pair_from_f4(const float4 p0, const float4 p1,
                                                   const float4 p2, const float4 g0,
                                                   const float4 g1, const float4 g2) {
  const float pa[6] = {p0.x, p0.y, p0.z, p0.w, p1.x, p1.y};
  const float pb[6] = {p1.z, p1.w, p2.x, p2.y, p2.z, p2.w};
  const float ga[6] = {g0.x, g0.y, g0.z, g0.w, g1.x, g1.y};
  const float gb[6] = {g1.z, g1.w, g2.x, g2.y, g2.z, g2.w};
  return pose_item(pa, ga) + pose_item(pb, gb);
}

// ---- stage 1: async double-buffered grid-stride over pairs -----------------
__global__ void __launch_bounds__(PL_THREADS)
pose_loss_stage1(const float* __restrict__ pred,
                 const float* __restrict__ gt,
                 float* __restrict__ partial, int npairs) {
  // Double buffer: per thread 96B (3x f4 pred + 3x f4 gt) per buffer = 48KB.
  __shared__ float4 stage[2][PL_THREADS * 6];

  const int t      = threadIdx.x;
  const int tidg   = blockIdx.x * PL_THREADS + t;
  const int stride = gridDim.x * PL_THREADS;

  // LDS byte offsets of this thread's slots in each buffer.
  const unsigned lds0 = (unsigned)(unsigned long long)
      (__attribute__((address_space(3))) float4*)&stage[0][t * 6];
  const unsigned lds1 = (unsigned)(unsigned long long)
      (__attribute__((address_space(3))) float4*)&stage[1][t * 6];

  float acc = 0.0f;

  // Wave-uniform number of full async tiles (no divergence in issue/wait).
  const int niter = npairs / stride;

  if (niter > 0) issue_tile_async(pred, gt, tidg, lds0);

  for (int it = 0; it < niter; ++it) {
    const unsigned nextbase = ((it + 1) & 1) ? lds1 : lds0;
    if (it + 1 < niter)
      issue_tile_async(pred, gt, tidg + (it + 1) * stride, nextbase);

    // One tile = 6 async b128s/wave; completion is in-order, so cnt<=6
    // releases the previous tile while the next stays in flight.
    if (it + 1 < niter) asm volatile("s_wait_asynccnt 6" ::: "memory");
    else                asm volatile("s_wait_asynccnt 0" ::: "memory");

    const float4* cur = &stage[it & 1][t * 6];
    acc += pose_pair_from_f4(cur[0], cur[1], cur[2], cur[3], cur[4], cur[5]);
  }

  // Remainder pairs (only when gridDim was clamped): plain b128 loads.
  for (int j = niter * stride + tidg; j < npairs; j += stride) {
    const float4* p4 = (const float4*)(pred + (size_t)j * 12);
    const float4* g4 = (const float4*)(gt + (size_t)j * 12);
    acc += pose_pair_from_f4(p4[0], p4[1], p4[2], g4[0], g4[1], g4[2]);
  }

  // wave32 butterfly reduction
#pragma unroll
  for (int off = 16; off > 0; off >>= 1) acc += __shfl_xor(acc, off, 32);

  __shared__ float smem[PL_THREADS / 32];
  const int lane = threadIdx.x & 31;
  const int wave = threadIdx.x >> 5;
  if (lane == 0) smem[wave] = acc;
  __syncthreads();
  if (threadIdx.x == 0) {
    float s = 0.0f;
#pragma unroll
    for (int i = 0; i < PL_THREADS / 32; ++i) s += smem[i];
    partial[blockIdx.x] = s;
  }
}

// ---- stage 2: fold block partials, scale by 1/N ----------------------------
__global__ void __launch_bounds__(PL_THREADS)
pose_loss_stage2(const float* __restrict__ partial, float* __restrict__ out,
                 int nblocks, float invN) {
  float acc = 0.0f;
  for (int i = threadIdx.x; i < nblocks; i += blockDim.x) acc += partial[i];
#pragma unroll
  for (int off = 16; off > 0; off >>= 1) acc += __shfl_xor(acc, off, 32);

  __shared__ float smem[PL_THREADS / 32];
  const int lane = threadIdx.x & 31;
  const int wave = threadIdx.x >> 5;
  if (lane == 0) smem[wave] = acc;
  __syncthreads();
  if (threadIdx.x == 0) {
    float s = 0.0f;
#pragma unroll
    for (int i = 0; i < PL_THREADS / 32; ++i) s += smem[i];
    out[0] = s * invN;
  }
}

// ---- entry -----------------------------------------------------------------
extern "C" void kernel_launch(void* const* d_in, const int* in_sizes, int n_in,
                              void* d_out, int out_size, void* d_ws,
                              size_t ws_size, hipStream_t stream) {
  const float* pred = (const float*)d_in[0];
  const float* gt   = (const float*)d_in[1];
  const int n       = in_sizes[0] / 6;   // N poses
  const int npairs  = n >> 1;            // N is even (2^21)

  int nblk = PL_BLOCKS;
  if (ws_size < (size_t)nblk * sizeof(float))
    nblk = (int)(ws_size / sizeof(float));
  if (nblk < 1) nblk = 1;

  float* partial = (float*)d_ws;

  pose_loss_stage1<<<nblk, PL_THREADS, 0, stream>>>(pred, gt, partial, npairs);
  pose_loss_stage2<<<1, PL_THREADS, 0, stream>>>(partial, (float*)d_out, nblk,
                                                 1.0f / (float)n);
}